// MetricPoseLoss_84361747628201
// MI455X (gfx1250) — compile-verified
//
#include <hip/hip_runtime.h>
#include <math.h>

typedef float v2f __attribute__((ext_vector_type(2)));
typedef float v8f __attribute__((ext_vector_type(8)));

#define B_ 8
#define NK_ 1024
#define S_ 256
#define IT_M_ 16
#define IT_R_ 16
#define NC_ 5
#define TH_ 0.15f
#define TEMP_ 10.0f
#define NREF_ 4
#define BETA_ 5.0f
#define MAX_ROT_ 45.0f
#define MAX_TRANS_ 1.0f
#define BV_ 128          // B * IT_M
#define SP 258           // padded LDS row stride (floats): bank-spread + 8B alignment

// ---------- 3x3 inverse (adjugate) ----------
__device__ inline void inv3(const float* __restrict__ K, float* Ki) {
  float a=K[0], b=K[1], c=K[2];
  float d=K[3], e=K[4], f=K[5];
  float g=K[6], h=K[7], i=K[8];
  float c00 =  (e*i - f*h);
  float c01 = -(d*i - f*g);
  float c02 =  (d*h - e*g);
  float inv = 1.0f / (a*c00 + b*c01 + c*c02);
  Ki[0] = c00*inv;  Ki[1] = (c*h - b*i)*inv;  Ki[2] = (b*f - c*e)*inv;
  Ki[3] = c01*inv;  Ki[4] = (a*i - c*g)*inv;  Ki[5] = (c*d - a*f)*inv;
  Ki[6] = c02*inv;  Ki[7] = (b*g - a*h)*inv;  Ki[8] = (a*e - b*d)*inv;
}

// ---------- 3x3 SVD (one-sided Jacobi) -> Kabsch rotation ----------
// R = V * diag(1,1,det(V)det(U)) * U^T, singular values sorted descending
__device__ void svd_rot(const float* Hm, float* R) {
  float A[9], V[9] = {1.f,0.f,0.f, 0.f,1.f,0.f, 0.f,0.f,1.f};
#pragma unroll
  for (int k = 0; k < 9; ++k) A[k] = Hm[k];
  for (int sweep = 0; sweep < 12; ++sweep) {
#pragma unroll
    for (int pq = 0; pq < 3; ++pq) {
      const int p = (pq < 2) ? 0 : 1;
      const int q = (pq == 0) ? 1 : 2;
      float al = 0.f, be = 0.f, ga = 0.f;
#pragma unroll
      for (int r = 0; r < 3; ++r) {
        al += A[r*3+p]*A[r*3+p];
        be += A[r*3+q]*A[r*3+q];
        ga += A[r*3+p]*A[r*3+q];
      }
      if (fabsf(ga) > 1e-18f) {
        float zeta = (be - al) / (2.0f * ga);
        float tt = copysignf(1.0f, zeta) / (fabsf(zeta) + sqrtf(1.0f + zeta*zeta));
        float cc = 1.0f / sqrtf(1.0f + tt*tt);
        float ss = cc * tt;
#pragma unroll
        for (int r = 0; r < 3; ++r) {
          float ap = cc*A[r*3+p] - ss*A[r*3+q];
          float aq = ss*A[r*3+p] + cc*A[r*3+q];
          A[r*3+p] = ap; A[r*3+q] = aq;
          float vp = cc*V[r*3+p] - ss*V[r*3+q];
          float vq = ss*V[r*3+p] + cc*V[r*3+q];
          V[r*3+p] = vp; V[r*3+q] = vq;
        }
      }
    }
  }
  float U[9], sg[3];
#pragma unroll
  for (int c = 0; c < 3; ++c) {
    float n = sqrtf(A[c]*A[c] + A[3+c]*A[3+c] + A[6+c]*A[6+c]);
    sg[c] = n;
    float in = 1.0f / (n + 1e-20f);
    U[c] = A[c]*in; U[3+c] = A[3+c]*in; U[6+c] = A[6+c]*in;
  }
  // sort columns by singular value, descending (det-flip must hit smallest)
#pragma unroll
  for (int pass = 0; pass < 3; ++pass) {
    int i = (pass == 0) ? 0 : (pass == 1 ? 0 : 1);
    int j = (pass == 0) ? 1 : 2;
    if (sg[i] < sg[j]) {
      float t0 = sg[i]; sg[i] = sg[j]; sg[j] = t0;
#pragma unroll
      for (int r = 0; r < 3; ++r) {
        float tu = U[r*3+i]; U[r*3+i] = U[r*3+j]; U[r*3+j] = tu;
        float tv = V[r*3+i]; V[r*3+i] = V[r*3+j]; V[r*3+j] = tv;
      }
    }
  }
  float detU = U[0]*(U[4]*U[8]-U[5]*U[7]) - U[1]*(U[3]*U[8]-U[5]*U[6]) + U[2]*(U[3]*U[7]-U[4]*U[6]);
  float detV = V[0]*(V[4]*V[8]-V[5]*V[7]) - V[1]*(V[3]*V[8]-V[5]*V[6]) + V[2]*(V[3]*V[7]-V[4]*V[6]);
  float dd = detU * detV;
#pragma unroll
  for (int i = 0; i < 3; ++i)
#pragma unroll
    for (int j = 0; j < 3; ++j)
      R[i*3+j] = V[i*3+0]*U[j*3+0] + V[i*3+1]*U[j*3+1] + dd*V[i*3+2]*U[j*3+2];
}

// =====================================================================
// Main kernel: one block per bv row (128 blocks, 128 threads = 4 waves).
// Each wave owns 4 hypotheses; WMMA f32 16x16x4 computes all 16 weighted
// covariance sums per hypothesis (4x4 diagonal blocks of the 16x16 acc).
// =====================================================================
__global__ __launch_bounds__(128) void pose_kernel(
    const float* __restrict__ kps0, const float* __restrict__ kps1,
    const float* __restrict__ depth0, const float* __restrict__ depth1,
    const float* __restrict__ T01, const float* __restrict__ K0,
    const float* __restrict__ K1, const int* __restrict__ sidx,
    const int* __restrict__ sidx_r, float* __restrict__ loss_ws)
{
  __shared__ float sXT[4*SP];       // SoA: [X0 X1 X2 1] rows, per point
  __shared__ float sYT[4*SP];       // SoA: [Y0 Y1 Y2 1]
  __shared__ float sWcur[16*SP];    // current inlier weights per hypothesis
  __shared__ float sWfin[16*SP];    // best-so-far inlier weights
  __shared__ float sRef[16*SP];     // candidate inlier mask this round
  __shared__ float sDt[4*256];      // per-wave 16x16 WMMA accumulator spill
  __shared__ float sRm[16*9];
  __shared__ float sTv[16*3];
  __shared__ float sPart[16*8];
  __shared__ float sCnt[16], sPre[16];
  __shared__ float sScore[16], sRotL[16], sTransL[16];

  const int bv = blockIdx.x;        // 0..127
  const int b  = bv >> 4;           // batch
  const int t  = threadIdx.x;       // 0..127

  // ---------------- gather + backproject into LDS (SoA) ----------------
  float Ki0[9], Ki1[9];
  inv3(K0 + b*9, Ki0);
  inv3(K1 + b*9, Ki1);
  for (int s = t; s < S_; s += 128) {
    int idx = sidx[bv*S_ + s];
    int i0 = idx / NK_;
    int i1 = idx - i0 * NK_;
    {
      float u = kps0[(b*2+0)*NK_ + i0];
      float v = kps0[(b*2+1)*NK_ + i0];
      float d = depth0[b*NK_ + i0];
      sXT[0*SP+s] = d*(Ki0[0]*u + Ki0[1]*v + Ki0[2]);
      sXT[1*SP+s] = d*(Ki0[3]*u + Ki0[4]*v + Ki0[5]);
      sXT[2*SP+s] = d*(Ki0[6]*u + Ki0[7]*v + Ki0[8]);
      sXT[3*SP+s] = 1.0f;
    }
    {
      float u = kps1[(b*2+0)*NK_ + i1];
      float v = kps1[(b*2+1)*NK_ + i1];
      float d = depth1[b*NK_ + i1];
      sYT[0*SP+s] = d*(Ki1[0]*u + Ki1[1]*v + Ki1[2]);
      sYT[1*SP+s] = d*(Ki1[3]*u + Ki1[4]*v + Ki1[5]);
      sYT[2*SP+s] = d*(Ki1[6]*u + Ki1[7]*v + Ki1[8]);
      sYT[3*SP+s] = 1.0f;
    }
  }
  // ---------------- initial one-hot inlier weights ----------------
  for (int k = t; k < 16*SP; k += 128) { sWcur[k] = 0.0f; sWfin[k] = 0.0f; }
  __syncthreads();
  if (t < 16) {
    int n = bv * IT_R_ + t;
    for (int c = 0; c < NC_; ++c) {
      int j = sidx_r[n*NC_ + c];
      sWcur[t*SP + j] = 1.0f;
      sWfin[t*SP + j] = 1.0f;
    }
    sPre[t] = (float)NC_;
  }
  __syncthreads();

  // ---- weighted covariance via WMMA + Procrustes solve (all threads) ----
  auto cov_and_solve = [&](const float* __restrict__ wbuf) {
    const int lane = t & 31;
    const int wv   = t >> 5;        // wave id: hypotheses 4*wv .. 4*wv+3
    const int half = lane >> 4;     // K-pair selector (0 -> k0/k1, 1 -> k2/k3)
    const int l    = lane & 15;     // M (A side) == N (B side) index
    const float* wrow = wbuf + (4*wv + (l >> 2)) * SP;  // hypothesis weights
    const float* prow = sXT + (l & 3) * SP;             // component of [X;1]
    const float* qrow = sYT + (l & 3) * SP;             // component of [Y;1]
#if defined(__has_builtin) && __has_builtin(__builtin_amdgcn_wmma_f32_16x16x4_f32)
    v8f acc = {0.f,0.f,0.f,0.f,0.f,0.f,0.f,0.f};
    for (int kb = 0; kb < 64; ++kb) {
      const int s = kb*4 + half*2;
      v2f a, bb;
      a.x  = wrow[s]   * prow[s];      // A[m][k]   = w_h * P_i  (VGPR0: k even)
      a.y  = wrow[s+1] * prow[s+1];    // A[m][k+1]               (VGPR1: k odd)
      bb.x = qrow[s];                  // B[k][n]   = Q_j
      bb.y = qrow[s+1];
      acc = __builtin_amdgcn_wmma_f32_16x16x4_f32(
          false, a, false, bb, (short)0, acc, false, false);
    }
    // D layout: VGPR r -> M = r + 8*half, N = l
#pragma unroll
    for (int r = 0; r < 8; ++r)
      sDt[wv*256 + (r + half*8)*16 + l] = acc[r];
#else
    // VALU fallback (only if the f32 WMMA builtin is absent)
    if ((t & 7) == 0) {
      int h = t >> 3;
      float sm[16];
#pragma unroll
      for (int k = 0; k < 16; ++k) sm[k] = 0.f;
      for (int s = 0; s < S_; ++s) {
        float w = wbuf[h*SP + s];
#pragma unroll
        for (int i = 0; i < 4; ++i)
#pragma unroll
          for (int j = 0; j < 4; ++j)
            sm[i*4+j] += w * sXT[i*SP+s] * sYT[j*SP+s];
      }
      int wvh = h >> 2, hl = h & 3;
#pragma unroll
      for (int i = 0; i < 4; ++i)
#pragma unroll
        for (int j = 0; j < 4; ++j)
          sDt[wvh*256 + (4*hl+i)*16 + (4*hl+j)] = sm[i*4+j];
    }
#endif
    __syncthreads();
    if ((t & 7) == 0) {               // 16 owner threads: one hypothesis each
      const int h = t >> 3;
      const int wvh = h >> 2, hl = h & 3;
      const float* dt = sDt + wvh*256;
      float sm[16];
#pragma unroll
      for (int i = 0; i < 4; ++i)
#pragma unroll
        for (int j = 0; j < 4; ++j)
          sm[i*4+j] = dt[(4*hl+i)*16 + (4*hl+j)];
      float isw = 1.0f / (sm[15] + 1e-8f);
      float cx[3], cy[3], Hm[9];
#pragma unroll
      for (int i = 0; i < 3; ++i) { cx[i] = sm[i*4+3]*isw; cy[i] = sm[12+i]*isw; }
#pragma unroll
      for (int i = 0; i < 3; ++i)
#pragma unroll
        for (int j = 0; j < 3; ++j)
          Hm[i*3+j] = sm[i*4+j] - sm[i*4+3]*sm[12+j]*isw;   // H = M - sx*sy^T/sw
      float R[9];
      svd_rot(Hm, R);
#pragma unroll
      for (int i = 0; i < 3; ++i) {
        sTv[h*3+i] = cy[i] - (R[i*3+0]*cx[0] + R[i*3+1]*cx[1] + R[i*3+2]*cx[2]);
#pragma unroll
        for (int j = 0; j < 3; ++j) sRm[h*9+i*3+j] = R[i*3+j];
      }
    }
    __syncthreads();
  };

  // ---------------- NREF refinement rounds ----------------
  for (int round = 0; round < NREF_; ++round) {
    cov_and_solve(sWcur);
    const int h = t >> 3, l8 = t & 7;
    float R[9], tv[3];
#pragma unroll
    for (int k = 0; k < 9; ++k) R[k] = sRm[h*9+k];
#pragma unroll
    for (int k = 0; k < 3; ++k) tv[k] = sTv[h*3+k];
    float cnt = 0.f;
    for (int s = l8; s < S_; s += 8) {
      float x0 = sXT[s], x1 = sXT[SP+s], x2 = sXT[2*SP+s];
      float e0 = sYT[s]      - (R[0]*x0 + R[1]*x1 + R[2]*x2 + tv[0]);
      float e1 = sYT[SP+s]   - (R[3]*x0 + R[4]*x1 + R[5]*x2 + tv[1]);
      float e2 = sYT[2*SP+s] - (R[6]*x0 + R[7]*x1 + R[8]*x2 + tv[2]);
      float rr = sqrtf(e0*e0 + e1*e1 + e2*e2);
      float rf = (rr < TH_) ? 1.0f : 0.0f;
      sRef[h*SP+s] = rf;
      cnt += rf;
    }
    sPart[t] = cnt;
    __syncthreads();
    if (l8 == 0) {                    // deterministic fixed-order reduction
      float c = 0.f;
#pragma unroll
      for (int k = 0; k < 8; ++k) c += sPart[h*8+k];
      sCnt[h] = c;
    }
    __syncthreads();
    const bool imp = sCnt[h] > sPre[h];
    __syncthreads();
    if (imp) {
      for (int s = l8; s < S_; s += 8) {
        sWfin[h*SP+s] = sWcur[h*SP+s];  // keep weights that *produced* this R
        sWcur[h*SP+s] = sRef[h*SP+s];   // refine for next round
      }
      if (l8 == 0) sPre[h] = sCnt[h];
    }
    __syncthreads();
  }

  // ---------------- final solve + scoring ----------------
  cov_and_solve(sWfin);
  if ((t & 7) == 0) {
    const int h = t >> 3;
    const float* Tb = T01 + b*16;
    float tr = 0.f;
#pragma unroll
    for (int i = 0; i < 3; ++i)
#pragma unroll
      for (int j = 0; j < 3; ++j)
        tr += sRm[h*9+i*3+j] * Tb[i*4+j];
    float ca = 0.5f * (tr - 1.0f);
    ca = fminf(fmaxf(ca, -1.0f + 1e-6f), 1.0f - 1e-6f);
    float rot = acosf(ca) * 57.29577951308232f;      // degrees
    float d0 = sTv[h*3+0] - Tb[3];
    float d1 = sTv[h*3+1] - Tb[7];
    float d2 = sTv[h*3+2] - Tb[11];
    float te = sqrtf(d0*d0 + d1*d1 + d2*d2);
    sRotL[h]   = MAX_ROT_   * tanhf(rot / MAX_ROT_);
    sTransL[h] = MAX_TRANS_ * tanhf(te  / MAX_TRANS_);
  }
  {
    const int h = t >> 3, l8 = t & 7;
    float R[9], tv[3];
#pragma unroll
    for (int k = 0; k < 9; ++k) R[k] = sRm[h*9+k];
#pragma unroll
    for (int k = 0; k < 3; ++k) tv[k] = sTv[h*3+k];
    float sc = 0.f;
    for (int s = l8; s < S_; s += 8) {
      float x0 = sXT[s], x1 = sXT[SP+s], x2 = sXT[2*SP+s];
      float e0 = sYT[s]      - (R[0]*x0 + R[1]*x1 + R[2]*x2 + tv[0]);
      float e1 = sYT[SP+s]   - (R[3]*x0 + R[4]*x1 + R[5]*x2 + tv[1]);
      float e2 = sYT[2*SP+s] - (R[6]*x0 + R[7]*x1 + R[8]*x2 + tv[2]);
      float rr = sqrtf(e0*e0 + e1*e1 + e2*e2);
      sc += 1.0f / (1.0f + expf(-BETA_ * (1.0f - rr / TH_)));
    }
    sPart[t] = sc;
    __syncthreads();
    if (l8 == 0) {
      float c = 0.f;
#pragma unroll
      for (int k = 0; k < 8; ++k) c += sPart[h*8+k];
      sScore[h] = c;
    }
  }
  __syncthreads();
  if (t == 0) {                       // softmax over IT_R, weighted losses
    float mx = -1e30f;
    for (int h = 0; h < 16; ++h) mx = fmaxf(mx, sScore[h] / TEMP_);
    float e[16], sum = 0.f;
    for (int h = 0; h < 16; ++h) { e[h] = expf(sScore[h]/TEMP_ - mx); sum += e[h]; }
    float inv = 1.0f / sum, lr = 0.f, lt = 0.f;
    for (int h = 0; h < 16; ++h) {
      float w = e[h] * inv;
      lr += sRotL[h] * w;
      lt += sTransL[h] * w;
    }
    loss_ws[bv]       = lr;
    loss_ws[BV_ + bv] = lt;
  }
}

// Deterministic final reduction: mean over IT_M per batch; out = [rot(8), trans(8)]
__global__ void reduce_kernel(const float* __restrict__ ws, float* __restrict__ out) {
  int t = threadIdx.x;
  if (t < 16) {
    const float* src = (t < 8) ? (ws + t*IT_M_) : (ws + BV_ + (t-8)*IT_M_);
    float a = 0.f;
#pragma unroll
    for (int m = 0; m < IT_M_; ++m) a += src[m];
    out[t] = a * (1.0f / (float)IT_M_);
  }
}

extern "C" void kernel_launch(void* const* d_in, const int* in_sizes, int n_in,
                              void* d_out, int out_size, void* d_ws, size_t ws_size,
                              hipStream_t stream) {
  (void)in_sizes; (void)n_in; (void)out_size; (void)ws_size;
  const float* kps0   = (const float*)d_in[1];
  const float* kps1   = (const float*)d_in[2];
  const float* depth0 = (const float*)d_in[3];
  const float* depth1 = (const float*)d_in[4];
  const float* T01    = (const float*)d_in[5];
  const float* K0     = (const float*)d_in[6];
  const float* K1     = (const float*)d_in[7];
  const int*   sidx   = (const int*)d_in[10];
  const int*   sidx_r = (const int*)d_in[11];
  float* ws  = (float*)d_ws;
  float* out = (float*)d_out;

  hipLaunchKernelGGL(pose_kernel, dim3(BV_), dim3(128), 0, stream,
                     kps0, kps1, depth0, depth1, T01, K0, K1, sidx, sidx_r, ws);
  hipLaunchKernelGGL(reduce_kernel, dim3(1), dim3(32), 0, stream, ws, out);
}